// DilateLoss_33621003993741
// MI455X (gfx1250) — compile-verified
//
#include <hip/hip_runtime.h>
#include <hip/hip_bf16.h>

#define NDIM 512
#define NBATCH 64
#define ALPHA_C 0.5f
#define GAMMA_C 0.01f
#define BIGV 1e10f
#define NEGBIG (-1e30f)

typedef __attribute__((ext_vector_type(2))) float v2f;
typedef __attribute__((ext_vector_type(8))) float v8f;

#define AS1 __attribute__((address_space(1)))
#define AS3 __attribute__((address_space(3)))

// ---------------------------------------------------------------------------
// One block per batch element. threadIdx.x = row i. Forward soft-DTW wavefront
// (storing R to global scratch), then backward soft-alignment wavefront
// accumulating the Omega-weighted path sum. Deterministic (no atomics).
// ---------------------------------------------------------------------------
__global__ __launch_bounds__(NDIM) void dilate_dp_kernel(
    const float* __restrict__ outputs,   // o, indexed by column j
    const float* __restrict__ targets,   // t, indexed by row i
    float* __restrict__ Rws,             // [NBATCH][N][N] scratch
    float* __restrict__ vals,            // [NBATCH] soft-DTW values
    float* __restrict__ tsums)           // [NBATCH] sum_ij E*(i-j)^2
{
  const int N = NDIM;
  const int k = blockIdx.x;
  const int i = threadIdx.x;

  __shared__ float lds_t[NDIM];
  __shared__ float lds_o[NDIM];
  __shared__ float bufR[3][NDIM];   // rotating R diagonals
  __shared__ float bufE[3][NDIM];   // rotating E diagonals (backward)
  __shared__ float red[NDIM];       // reduction scratch

  const float* gt = targets + (size_t)k * N;
  const float* go = outputs + (size_t)k * N;

  // ---- stage t/o into LDS via CDNA5 async global->LDS (ASYNCcnt path) ----
#if defined(__gfx1250__) && __has_builtin(__builtin_amdgcn_global_load_async_to_lds_b32)
  {
    AS1 int* g0 = (AS1 int*)(unsigned long long)(size_t)(gt + i);
    AS1 int* g1 = (AS1 int*)(unsigned long long)(size_t)(go + i);
    AS3 int* l0 = (AS3 int*)(unsigned int)(size_t)(&lds_t[i]);
    AS3 int* l1 = (AS3 int*)(unsigned int)(size_t)(&lds_o[i]);
    __builtin_amdgcn_global_load_async_to_lds_b32(g0, l0, 0, 0);
    __builtin_amdgcn_global_load_async_to_lds_b32(g1, l1, 0, 0);
#if __has_builtin(__builtin_amdgcn_s_wait_asynccnt)
    __builtin_amdgcn_s_wait_asynccnt(0);
#else
    asm volatile("s_wait_asynccnt 0" ::: "memory");
#endif
  }
#else
  lds_t[i] = gt[i];
  lds_o[i] = go[i];
#endif
  bufR[0][i] = BIGV; bufR[1][i] = BIGV; bufR[2][i] = BIGV;
  __syncthreads();

  const float ti   = lds_t[i];
  const float invg = 1.0f / GAMMA_C;
  float* Rk = Rws + (size_t)k * N * N;

  // ------------------------------ forward ------------------------------
  // R[i,j] = D[i,j] + softmin_g(R[i-1,j], R[i-1,j-1], R[i,j-1]); R[-1,-1]=0
  for (int d = 0; d < 2 * N - 1; ++d) {
    const int j  = d - i;
    const int b0 = d % 3;          // diag d   (write)
    const int b1 = (d + 2) % 3;    // diag d-1
    const int b2 = (d + 1) % 3;    // diag d-2
    float r0 = BIGV;
    if (j >= 0 && j < N) {
      float a = (i >= 1) ? bufR[b1][i - 1] : BIGV;                       // R[i-1,j]
      float b = (i >= 1) ? bufR[b2][i - 1] : ((d == 0) ? 0.0f : BIGV);   // R[i-1,j-1]
      float c = bufR[b1][i];                                             // R[i,j-1]
      float m = fminf(a, fminf(b, c));
      float s = __expf((m - a) * invg) + __expf((m - b) * invg) + __expf((m - c) * invg);
      float smin = m - GAMMA_C * __logf(s);
      float diff = ti - lds_o[j];
      r0 = __fmaf_rn(diff, diff, smin);
      Rk[(size_t)i * N + j] = r0;
    }
    bufR[b0][i] = r0;
    __syncthreads();
  }

  if (i == 0) vals[k] = bufR[(2 * N - 2) % 3][N - 1];   // R[N-1,N-1]
  __syncthreads();

  // ------------------------------ backward ------------------------------
  // E[i,j] = sum over children (i+1,j),(i,j+1),(i+1,j+1) of
  //          exp((R[child]-R[i,j]-D[child])/g) * E[child];  E[N-1,N-1]=1
  bufE[0][i] = 0.0f; bufE[1][i] = 0.0f; bufE[2][i] = 0.0f;
  bufR[0][i] = NEGBIG; bufR[1][i] = NEGBIG; bufR[2][i] = NEGBIG;
  __syncthreads();

  float tacc = 0.0f;
  for (int d = 2 * N - 2; d >= 0; --d) {
    const int j   = d - i;
    const int b0  = d % 3;          // diag d   (write)
    const int bp1 = (d + 1) % 3;    // diag d+1
    const int bp2 = (d + 2) % 3;    // diag d+2
    float rij = NEGBIG, eij = 0.0f;
    if (j >= 0 && j < N) {
      rij = Rk[(size_t)i * N + j];
      if (i == N - 1 && j == N - 1) {
        eij = 1.0f;
      } else {
        float acc = 0.0f;
        if (i + 1 < N) {                       // child (i+1, j)
          float rn = bufR[bp1][i + 1];
          float en = bufE[bp1][i + 1];
          float dn = lds_t[i + 1] - lds_o[j];  dn *= dn;
          acc += __expf((rn - rij - dn) * invg) * en;
        }
        if (j + 1 < N) {                       // child (i, j+1)
          float rn = bufR[bp1][i];
          float en = bufE[bp1][i];
          float dn = ti - lds_o[j + 1];        dn *= dn;
          acc += __expf((rn - rij - dn) * invg) * en;
        }
        if (i + 1 < N && j + 1 < N) {          // child (i+1, j+1)
          float rn = bufR[bp2][i + 1];
          float en = bufE[bp2][i + 1];
          float dn = lds_t[i + 1] - lds_o[j + 1]; dn *= dn;
          acc += __expf((rn - rij - dn) * invg) * en;
        }
        eij = acc;
      }
      float dij = (float)(i - j);
      tacc = __fmaf_rn(eij, dij * dij, tacc);
    }
    bufE[b0][i] = eij;
    bufR[b0][i] = rij;
    __syncthreads();
  }

  // deterministic block tree-reduction of the temporal sum
  red[i] = tacc;
  __syncthreads();
  for (int s = NDIM / 2; s > 0; s >>= 1) {
    if (i < s) red[i] += red[i + s];
    __syncthreads();
  }
  if (i == 0) tsums[k] = red[0];
}

// ---------------------------------------------------------------------------
// Finalize: loss = alpha*mean(vals) + (1-alpha)*mean(tsums)/(N*N).
// The 64-element weighted batch reduction is done as a 16x4 WMMA row-sum
// against a ones matrix (v_wmma_f32_16x16x4_f32), full f32 precision.
// ---------------------------------------------------------------------------
__global__ __launch_bounds__(32) void dilate_finalize_kernel(
    const float* __restrict__ vals, const float* __restrict__ tsums,
    float* __restrict__ out)
{
  const int lane = threadIdx.x;
  const float wShape = ALPHA_C / (float)NBATCH;
  const float wTemp  = (1.0f - ALPHA_C) / ((float)NBATCH * (float)NDIM * (float)NDIM);
#if defined(__gfx1250__) && __has_builtin(__builtin_amdgcn_wmma_f32_16x16x4_f32)
  // A[M][K] = combined[4*M + K]; lanes 0-15 hold K=0,1, lanes 16-31 hold K=2,3
  const int M  = lane & 15;
  const int kk = 4 * M + ((lane >> 4) << 1);
  v2f a;
  a.x = wShape * vals[kk]     + wTemp * tsums[kk];
  a.y = wShape * vals[kk + 1] + wTemp * tsums[kk + 1];
  v2f b; b.x = 1.0f; b.y = 1.0f;   // ones B-matrix -> D[M][n] = rowsum_M
  v8f c = {};
  v8f dm = __builtin_amdgcn_wmma_f32_16x16x4_f32(
      /*neg_a=*/false, a, /*neg_b=*/false, b,
      /*c_mod=*/(short)0, c, /*reuse_a=*/false, /*reuse_b=*/false);
  float part = dm[0] + dm[1] + dm[2] + dm[3] + dm[4] + dm[5] + dm[6] + dm[7];
  float other = __shfl(part, lane ^ 16, 32);   // half-rows live 16 lanes apart
  if (lane == 0) out[0] = part + other;
#else
  if (lane == 0) {
    float accS = 0.0f, accT = 0.0f;
    for (int k = 0; k < NBATCH; ++k) { accS += vals[k]; accT += tsums[k]; }
    out[0] = wShape * accS + wTemp * accT;
  }
#endif
}

extern "C" void kernel_launch(void* const* d_in, const int* in_sizes, int n_in,
                              void* d_out, int out_size, void* d_ws, size_t ws_size,
                              hipStream_t stream)
{
  const float* outputs = (const float*)d_in[0];   // (B, N, 1) float32
  const float* targets = (const float*)d_in[1];   // (B, N, 1) float32

  float* Rws   = (float*)d_ws;                                         // 64 MB
  float* vals  = (float*)((char*)d_ws +
                          (size_t)NBATCH * NDIM * NDIM * sizeof(float));
  float* tsums = vals + NBATCH;

  dilate_dp_kernel<<<NBATCH, NDIM, 0, stream>>>(outputs, targets, Rws, vals, tsums);
  dilate_finalize_kernel<<<1, 32, 0, stream>>>(vals, tsums, (float*)d_out);

  (void)in_sizes; (void)n_in; (void)out_size; (void)ws_size;
}